// TopKRouter_21741124452485
// MI455X (gfx1250) — compile-verified
//
#include <hip/hip_runtime.h>
#include <hip/hip_bf16.h>

// TopK MoE router for MI455X (gfx1250, wave32, WMMA + async-to-LDS).
//
// logits = x @ W^T + b (M=16384 tokens, N=64 experts, K=4096, fp32), then
// per-token top-2 -> sparse softmax (non-top2 entries exactly 0.0).
//
// Roofline: 8.6 GFLOP vs 256 MB of x streamed once -> ~33 FLOP/byte ->
// HBM-bound at 23.3 TB/s (~11 us floor). Round-1 codegen stalled on
// `s_wait_loadcnt 0` before every WMMA (loads consumed immediately after
// issue). This version double-buffers K-chunks through LDS with
// GLOBAL_LOAD_ASYNC_TO_LDS_B128 (ASYNCcnt), so chunk c+1 streams from HBM
// while chunk c's 64 WMMAs run out of LDS. W chunks are staged once per
// block and shared by all 4 waves (4x less L2 traffic).

typedef __attribute__((ext_vector_type(2))) float v2f;
typedef __attribute__((ext_vector_type(4))) float v4f;
typedef __attribute__((ext_vector_type(8))) float v8f;

#define EXPERTS 64
#define TPW 16                 // tokens (M rows) per wave
#define WPB 4                  // waves per block
#define BLOCK (WPB * 32)
#define BK 64                  // K-chunk (floats)
#define PITCH 68               // LDS row pitch in floats: bank = (4*row + k) % 64
                               // -> conflict-free b64 fragment reads; 272B keeps
                               // 16B alignment for async B128 writes.

// Async global->LDS copy, 16B per lane. VDST = LDS byte offset (low 32 bits of
// a generic pointer to __shared__), VADDR = 64-bit global address.
__device__ __forceinline__ void async_b128(const void* g, void* l) {
    asm volatile("global_load_async_to_lds_b128 %0, %1, off"
                 :: "v"((unsigned)(size_t)l), "v"(g)
                 : "memory");
}

#if __has_builtin(__builtin_amdgcn_s_wait_asynccnt)
#define WAIT_ASYNC(n) __builtin_amdgcn_s_wait_asynccnt(n)
#else
#define WAIT_ASYNC(n) asm volatile("s_wait_asynccnt %0" :: "n"(n) : "memory")
#endif

__global__ __launch_bounds__(BLOCK) void topk_router_wmma(
    const float* __restrict__ x,
    const float* __restrict__ W,
    const float* __restrict__ bias,
    float* __restrict__ out_probs,   // [T, 64]
    float* __restrict__ out_idx,     // [T, 2] (indices value-cast to float)
    int D)
{
    __shared__ __align__(16) float xs[2][WPB][TPW][PITCH];      // 34.0 KB
    __shared__ __align__(16) float ws[2][EXPERTS][PITCH];       // 34.0 KB
    __shared__ float sm[WPB][TPW][EXPERTS + 1];                 // 16.3 KB

    const int lane = threadIdx.x & 31;
    const int wave = threadIdx.x >> 5;
    const int mrow = lane & 15;      // M row / N col / token-in-tile
    const int hi   = lane >> 4;      // half-wave select

    const int tile_m = (blockIdx.x * WPB + wave) * TPW;
    const int scol   = mrow * 4;     // staging column (floats); 16 lanes x 16B = 1 row

    const float* xbase = x + (size_t)tile_m * D;          // this wave's 16 tokens
    const float* wbase = W + (size_t)(wave * TPW) * D;    // this wave's 16 experts
    const int NC = D / BK;

    // ---- stage chunk 0 (16 async B128 ops per wave: 8 for x, 8 for W share) ----
    #pragma unroll
    for (int i = 0; i < 8; ++i) {
        const int r = 2 * i + hi;    // 2 rows per op (lanes 0-15 / 16-31)
        async_b128(xbase + (size_t)r * D + scol, &xs[0][wave][r][scol]);
    }
    #pragma unroll
    for (int i = 0; i < 8; ++i) {
        const int r = 2 * i + hi;
        async_b128(wbase + (size_t)r * D + scol, &ws[0][wave * TPW + r][scol]);
    }

    v8f acc0 = {0.f,0.f,0.f,0.f,0.f,0.f,0.f,0.f};
    v8f acc1 = acc0, acc2 = acc0, acc3 = acc0;

    for (int c = 0; c < NC; ++c) {
        const int b = c & 1;

        if (c + 1 < NC) {
            // Prefetch chunk c+1 into the other buffer, then wait until only
            // those 16 just-issued ops are outstanding => chunk c is resident.
            const int nb = b ^ 1;
            const int k1 = (c + 1) * BK;
            #pragma unroll
            for (int i = 0; i < 8; ++i) {
                const int r = 2 * i + hi;
                async_b128(xbase + (size_t)r * D + k1 + scol, &xs[nb][wave][r][scol]);
            }
            #pragma unroll
            for (int i = 0; i < 8; ++i) {
                const int r = 2 * i + hi;
                async_b128(wbase + (size_t)r * D + k1 + scol,
                           &ws[nb][wave * TPW + r][scol]);
            }
            WAIT_ASYNC(16);
        } else {
            WAIT_ASYNC(0);
        }
        __syncthreads();   // all waves' staging of chunk c complete (incl. shared W)

        // A 16x4 f32 fragment: lanes 0-15 -> M=0..15, K={0,1}; lanes 16-31 ->
        // K={2,3}: per lane a contiguous float2.  B 4x16 assumed mirrored.
        #pragma unroll
        for (int k = 0; k < BK; k += 4) {
            const int kk = k + 2 * hi;
            v2f a  = *(const v2f*)&xs[b][wave][mrow][kk];
            v2f b0 = *(const v2f*)&ws[b][ 0 + mrow][kk];
            v2f b1 = *(const v2f*)&ws[b][16 + mrow][kk];
            v2f b2 = *(const v2f*)&ws[b][32 + mrow][kk];
            v2f b3 = *(const v2f*)&ws[b][48 + mrow][kk];
            acc0 = __builtin_amdgcn_wmma_f32_16x16x4_f32(false, a, false, b0,
                                                         (short)0, acc0, false, false);
            acc1 = __builtin_amdgcn_wmma_f32_16x16x4_f32(false, a, false, b1,
                                                         (short)0, acc1, false, false);
            acc2 = __builtin_amdgcn_wmma_f32_16x16x4_f32(false, a, false, b2,
                                                         (short)0, acc2, false, false);
            acc3 = __builtin_amdgcn_wmma_f32_16x16x4_f32(false, a, false, b3,
                                                         (short)0, acc3, false, false);
        }
        __syncthreads();   // chunk-c reads done block-wide before buffer b is
                           // overwritten by the stage of chunk c+2
    }

    // C/D layout: VGPR j, lanes 0-15 -> M=j, lanes 16-31 -> M=j+8; N = lane&15.
    const float bv0 = bias[ 0 + mrow];
    const float bv1 = bias[16 + mrow];
    const float bv2 = bias[32 + mrow];
    const float bv3 = bias[48 + mrow];
    const int mb = 8 * hi;
    #pragma unroll
    for (int j = 0; j < 8; ++j) {
        sm[wave][mb + j][ 0 + mrow] = acc0[j] + bv0;
        sm[wave][mb + j][16 + mrow] = acc1[j] + bv1;
        sm[wave][mb + j][32 + mrow] = acc2[j] + bv2;
        sm[wave][mb + j][48 + mrow] = acc3[j] + bv3;
    }
    __syncthreads();

    // Per-token top-2 (both half-waves scan the same token; each then writes
    // half of the 64-expert output row). Strict '>' matches jax first-
    // occurrence tie-breaking; results sorted descending.
    float best = -3.0e38f, sec = -3.0e38f;
    int bi = 0, si = 0;
    #pragma unroll
    for (int e = 0; e < EXPERTS; ++e) {
        float v = sm[wave][mrow][e];
        if (v > best)      { sec = best; si = bi; best = v; bi = e; }
        else if (v > sec)  { sec = v;    si = e; }
    }
    // Sparse softmax: non-top2 entries are exp(-1e30 - max) == 0.0 in fp32,
    // exactly matching the reference scatter+softmax.
    const float e1  = expf(sec - best);
    const float inv = 1.0f / (1.0f + e1);
    const float p0  = inv;
    const float p1  = e1 * inv;

    const size_t gtok = (size_t)tile_m + mrow;
    float* orow = out_probs + gtok * EXPERTS + hi * 32;
    const int eb = hi * 32;
    #pragma unroll
    for (int e4 = 0; e4 < 32; e4 += 4) {
        v4f v;
        #pragma unroll
        for (int cc = 0; cc < 4; ++cc) {
            const int e = eb + e4 + cc;
            v[cc] = (e == bi) ? p0 : ((e == si) ? p1 : 0.0f);
        }
        *(v4f*)(orow + e4) = v;
    }
    if (hi == 0) {
        out_idx[gtok * 2 + 0] = (float)bi;   // descending: best first
        out_idx[gtok * 2 + 1] = (float)si;
    }
}

extern "C" void kernel_launch(void* const* d_in, const int* in_sizes, int n_in,
                              void* d_out, int out_size, void* d_ws, size_t ws_size,
                              hipStream_t stream) {
    const float* x = (const float*)d_in[0];
    const float* W = (const float*)d_in[1];
    const float* b = (const float*)d_in[2];
    // d_in[3] = top_k scalar; kernel is specialized for top_k == 2.

    const int E = in_sizes[2];                       // 64 experts
    const int D = in_sizes[1] / E;                   // 4096
    const long long T = (long long)in_sizes[0] / D;  // 16384 tokens

    float* out_probs = (float*)d_out;                // T*E floats
    float* out_idx   = out_probs + (size_t)T * E;    // T*2 (cast indices)

    const int tokens_per_block = WPB * TPW;          // 64
    dim3 grid((unsigned)((T + tokens_per_block - 1) / tokens_per_block));
    topk_router_wmma<<<grid, BLOCK, 0, stream>>>(x, W, b, out_probs, out_idx, D);
}